// HashMapping_24043226923975
// MI455X (gfx1250) — compile-verified
//
#include <hip/hip_runtime.h>
#include <hip/hip_bf16.h>

// N-gram rolling-hash kernel for MI455X (gfx1250).
// Roofline: ~33.5 MB moved -> ~1.44 us at 23.3 TB/s HBM; VALU (~38 Mop int)
// is negligible, so the kernel is pure data movement.
// Path: CDNA5 async global->LDS staging (global_load_async_to_lds_b128,
// ASYNCcnt, s_wait_asynccnt), conflict-free LDS sliding-window reads,
// lane-coalesced b32 stores.

#define TILE     2048
#define THREADS  256
#define KPT      (TILE / THREADS)          // outputs per thread
#define HASH_MULT 1000003u
#define CAP_MASK  0xFFFFFu                 // CAPACITY = 1<<20 (power of two)

__global__ __launch_bounds__(THREADS)
void ngram_hash_kernel(const int* __restrict__ in,
                       const int* __restrict__ seedp,
                       const int* __restrict__ ngramp,
                       int* __restrict__ out,
                       int B, int L, int out_len) {
    __shared__ int smem[TILE + 64];        // tile + (n-1) tail + b128 round-up

    const int row      = blockIdx.y;
    const int tileBase = blockIdx.x * TILE;
    if (tileBase >= out_len) return;

    const unsigned n    = (unsigned)ngramp[0];   // static 4 in the reference
    const unsigned seed = (unsigned)seedp[0];

    const long long rowBase = (long long)row * (long long)L;
    const int avail = L - tileBase;              // ints left in this row
    int need = TILE + (int)n - 1;                // ints this tile consumes
    if (need > avail) need = avail;
    const int n4 = (need + 3) >> 2;              // b128 transfers (rounded up)
    const int* gsrc = in + rowBase + tileBase;

    // Async b128 path requires: 16B-aligned global addresses (L % 4 == 0 =>
    // every rowBase/tileBase is 16B aligned) and that the rounded-up reads
    // (up to +3 ints) never run past the end of the whole buffer (only
    // possible on the last row; interior rows just peek into the next row).
    const bool safe = ((L & 3) == 0) &&
                      ((row != B - 1) || (tileBase + 4 * n4 <= L));

    if (safe) {
        // CDNA5 async copy: each enabled lane DMAs 16B global -> LDS.
        // LDS byte address = low 32 bits of the generic shared pointer
        // (ISA 10.2: the LDS aperture tag lives entirely in addr[63:32]).
        for (int i = (int)threadIdx.x; i < n4; i += THREADS) {
            unsigned   ldsAddr = (unsigned)(unsigned long long)(&smem[i << 2]);
            const int* g       = gsrc + (i << 2);
            asm volatile("global_load_async_to_lds_b128 %0, %1, off"
                         :: "v"(ldsAddr), "v"(g)
                         : "memory");
        }
    } else {
        // Exact-bounds scalar staging (tail of last row / unaligned shapes).
        for (int i = (int)threadIdx.x; i < need; i += THREADS)
            smem[i] = gsrc[i];
    }
    __builtin_amdgcn_s_wait_asynccnt(0);   // drain this wave's async DMAs
    __syncthreads();                       // LDS staged block-wide

    int* __restrict__ orow = out + (long long)row * (long long)out_len + tileBase;
    const unsigned sM = seed * HASH_MULT;  // hoist first multiply (common seed)
    const bool fullTile = (tileBase + TILE <= out_len);

    if (n == 4u && fullTile) {
        // Hot path: 15/16 tiles. No bounds checks, unrolled window hash.
        // Lane-consecutive p -> conflict-free ds_load banks; store is
        // 128 contiguous bytes per wave instruction.
#pragma unroll
        for (int j = 0; j < KPT; ++j) {
            const int p = j * THREADS + (int)threadIdx.x;
            unsigned h = sM ^ (unsigned)smem[p];
            h = h * HASH_MULT ^ (unsigned)smem[p + 1];
            h = h * HASH_MULT ^ (unsigned)smem[p + 2];
            h = h * HASH_MULT ^ (unsigned)smem[p + 3];
            orow[p] = (int)(h & CAP_MASK);
        }
    } else if (n == 4u) {
        // Guarded tail tile (last tile of each row).
        const int lim = out_len - tileBase;
#pragma unroll
        for (int j = 0; j < KPT; ++j) {
            const int p = j * THREADS + (int)threadIdx.x;
            if (p < lim) {
                unsigned h = sM ^ (unsigned)smem[p];
                h = h * HASH_MULT ^ (unsigned)smem[p + 1];
                h = h * HASH_MULT ^ (unsigned)smem[p + 2];
                h = h * HASH_MULT ^ (unsigned)smem[p + 3];
                orow[p] = (int)(h & CAP_MASK);
            }
        }
    } else {
        // Generic ngram fallback (never taken for this reference).
        const int lim = out_len - tileBase;
#pragma unroll
        for (int j = 0; j < KPT; ++j) {
            const int p = j * THREADS + (int)threadIdx.x;
            if (p < lim) {
                unsigned h = seed;
                for (unsigned i = 0; i < n; ++i)
                    h = h * HASH_MULT ^ (unsigned)smem[p + (int)i];
                orow[p] = (int)(h & CAP_MASK);
            }
        }
    }
}

extern "C" void kernel_launch(void* const* d_in, const int* in_sizes, int n_in,
                              void* d_out, int out_size, void* d_ws, size_t ws_size,
                              hipStream_t stream) {
    (void)n_in; (void)d_ws; (void)ws_size;
    if (out_size <= 0) return;

    const int* in    = (const int*)d_in[0];   // [B, L] int32 byte values
    const int* seed  = (const int*)d_in[1];   // [1] int32
    const int* ngram = (const int*)d_in[2];   // [1] int32 (== 4)
    int*       out   = (int*)d_out;           // [B, L-n+1] int32

    // Recover shape from flat sizes. Reference's ngram_size is static 4:
    // in = B*L, out = B*(L-3)  =>  B = (in - out)/3.
    const long long total_in = (long long)in_sizes[0];
    int B = (int)((total_in - (long long)out_size) / 3);
    if (B <= 0) B = 1;
    const int L       = (int)(total_in / B);
    const int out_len = out_size / B;

    dim3 block(THREADS);
    dim3 grid((out_len + TILE - 1) / TILE, B);
    ngram_hash_kernel<<<grid, block, 0, stream>>>(in, seed, ngram, out,
                                                  B, L, out_len);
}